// GCN_48610439856190
// MI455X (gfx1250) — compile-verified
//
#include <hip/hip_runtime.h>
#include <hip/hip_bf16.h>

// ---------------------------------------------------------------------------
// GCN layer: out = relu( D^-1/2 (A_set + I) D^-1/2 (X @ W) )
//   N_NODES=10000, N_EDGES=320000, F=256
// ---------------------------------------------------------------------------

#define N_NODES 10000
#define N_EDGES 320000
#define FDIM    256
#define W32P    320      // mask row words, padded to 1280B (16B-aligned for b128)
#define XS_STRIDE 260    // LDS row stride in floats (256 + 4 pad -> bank-conflict-free)

typedef __attribute__((ext_vector_type(2))) float    v2f;
typedef __attribute__((ext_vector_type(8))) float    v8f;
typedef __attribute__((ext_vector_type(4))) unsigned uint4v;
typedef __attribute__((ext_vector_type(8))) int      int8v;
typedef __attribute__((ext_vector_type(4))) int      int4v;

typedef __attribute__((address_space(3))) void  lds_v;   // LDS void
typedef __attribute__((address_space(1))) int4v gint4;   // global v4i
typedef __attribute__((address_space(3))) int4v lint4;   // LDS v4i

// ---- CDNA5 data-mover feature detection (compile-safe fallbacks) ----------
#if defined(__has_builtin)
#  if __has_builtin(__builtin_amdgcn_tensor_load_to_lds)
#    define HAVE_TDM 1
#  endif
#  if __has_builtin(__builtin_amdgcn_global_load_async_to_lds_b128)
#    define HAVE_ASYNC 1
#  endif
#endif

#if defined(__has_builtin) && __has_builtin(__builtin_amdgcn_s_wait_asynccnt)
#  define WAIT_ASYNC0() __builtin_amdgcn_s_wait_asynccnt(0)
#else
#  define WAIT_ASYNC0() asm volatile("s_wait_asynccnt 0x0" ::: "memory")
#endif
#if defined(__has_builtin) && __has_builtin(__builtin_amdgcn_s_wait_tensorcnt)
#  define WAIT_TENSOR0() __builtin_amdgcn_s_wait_tensorcnt(0)
#else
#  define WAIT_TENSOR0() asm volatile("s_wait_tensorcnt 0x0" ::: "memory")
#endif

// --- 1) build dedup'd adjacency bitmask + in-degree ------------------------
__global__ void gcn_build_adj(const int* __restrict__ src,
                              const int* __restrict__ tgt,
                              unsigned* __restrict__ mask,
                              unsigned* __restrict__ deg) {
    int e = blockIdx.x * blockDim.x + threadIdx.x;
    if (e >= N_EDGES) return;
    int s = src[e];
    int t = tgt[e];
    unsigned bit = 1u << (t & 31);
    unsigned old = atomicOr(&mask[(size_t)s * W32P + (t >> 5)], bit);
    if (!(old & bit)) atomicAdd(&deg[t], 1u);      // set-semantics degree
}

// --- 2) d^-1/2 (self-loop adds +1 to every column sum) ---------------------
__global__ void gcn_dinv(const unsigned* __restrict__ deg,
                         float* __restrict__ dinv) {
    int j = blockIdx.x * blockDim.x + threadIdx.x;
    if (j >= N_NODES) return;
    dinv[j] = rsqrtf((float)deg[j] + 1.0f);
}

// --- 3) H = X @ W  (fp32 WMMA 16x16x4) -------------------------------------
// Block = 512 thr (16 waves) owns a 16-row stripe; wave w computes col tile w.
// X stripe (16x256 f32) staged to LDS via TDM (padded stride 260) or async.
__global__ void __launch_bounds__(512)
gcn_gemm_wmma(const float* __restrict__ X,
              const float* __restrict__ W,
              float* __restrict__ H) {
    __shared__ float Xs[16 * XS_STRIDE];            // 16,640 B
    const int tid   = threadIdx.x;
    const int lane  = tid & 31;
    const int tileN = tid >> 5;                     // 0..15
    const int tileM = blockIdx.x;                   // 0..624
    const float* gX = X + (size_t)tileM * 16 * FDIM;

#if defined(HAVE_TDM)
    if (tid < 32) {                                 // one wave issues the DMA
        unsigned           ldsAddr = (unsigned)(unsigned long long)(lds_v*)Xs;
        unsigned long long ga      = (unsigned long long)(size_t)gX;
        // D# group0: count=1 | lds_addr | global_addr[56:0] | type=2
        uint4v g0 = { 1u, ldsAddr, (unsigned)ga,
                      (unsigned)((ga >> 32) & 0x01FFFFFFull) | 0x80000000u };
        // D# group1: data_size=4B(2), pad_enable, pad_interval=256dw(7),
        //            pad_amount=4dw(3); tensor 256x16, tile 256x16, stride0=256
        int8v g1 = { (int)0x07D20000,
                     (int)(256u << 16),             // tensor_dim0 = 256
                     (int)(16u  << 16),             // tensor_dim1 = 16
                     (int)(256u << 16),             // tile_dim0   = 256
                     (int)16,                       // tile_dim1   = 16
                     (int)256,                      // tensor_dim0_stride
                     0, 0 };
        int4v z4 = { 0, 0, 0, 0 };
        int8v z8 = { 0, 0, 0, 0, 0, 0, 0, 0 };
        __builtin_amdgcn_tensor_load_to_lds(g0, g1, z4, z4, z8, 0);
        WAIT_TENSOR0();
    }
#elif defined(HAVE_ASYNC)
    {   // each thread moves 8 floats = 2 x b128 (16B aligned both sides)
        const int r = tid >> 5;                     // 0..15
        const int c = (tid & 31) * 8;               // 0..248
        const float* g = gX + (size_t)r * FDIM + c;
        float*       l = &Xs[r * XS_STRIDE + c];
        __builtin_amdgcn_global_load_async_to_lds_b128(
            (gint4*)g,       (lint4*)l,       0, 0);
        __builtin_amdgcn_global_load_async_to_lds_b128(
            (gint4*)(g + 4), (lint4*)(l + 4), 0, 0);
        WAIT_ASYNC0();
    }
#else
    {
        const int r = tid >> 5;
        const int c = (tid & 31) * 8;
        #pragma unroll
        for (int k = 0; k < 8; ++k)
            Xs[r * XS_STRIDE + c + k] = gX[(size_t)r * FDIM + c + k];
    }
#endif
    __syncthreads();

    // A 16x4: lane(0-15)=M, vgprs=K pair; lanes 16-31 hold K{2,3}
    // B 4x16: lane(0-15)=N; C/D: vgpr r = row r / r+8, lane = N
    const int mrow  = lane & 15;
    const int n     = tileN * 16 + (lane & 15);
    const int khalf = (lane >> 4) << 1;             // 0 or 2

    v8f c = {};
    const float* xrow = &Xs[mrow * XS_STRIDE + khalf];
    const float* wcol = W + (size_t)khalf * FDIM + n;
    #pragma unroll 4
    for (int k0 = 0; k0 < FDIM; k0 += 4) {
        v2f a, b;
        a.x = xrow[k0];                             // ds_load_b64, conflict-free
        a.y = xrow[k0 + 1];
        b.x = wcol[(size_t)k0 * FDIM];
        b.y = wcol[(size_t)(k0 + 1) * FDIM];
        c = __builtin_amdgcn_wmma_f32_16x16x4_f32(
                false, a, false, b, (short)0, c, false, false);
    }

    const int rowBase = tileM * 16 + ((lane >> 4) << 3);
    #pragma unroll
    for (int r = 0; r < 8; ++r)
        H[(size_t)(rowBase + r) * FDIM + n] = c[r];
}

// --- 4) z[i] = relu( d_i * ( d_i*H[i] + sum_{j in adj(i)} d_j*H[j] ) ) -----
__global__ void gcn_aggregate(const float* __restrict__ H,
                              const unsigned* __restrict__ mask,
                              const float* __restrict__ dinv,
                              float* __restrict__ out) {
    __shared__ unsigned smask[W32P];                // 1280 B
    const int i = blockIdx.x;
    const int f = threadIdx.x;                      // 256 threads
#if defined(HAVE_ASYNC)
    if (f < W32P / 4) {                             // 80 threads x 16B
        const unsigned* g = mask + (size_t)i * W32P + f * 4;
        __builtin_amdgcn_global_load_async_to_lds_b128(
            (gint4*)g, (lint4*)&smask[f * 4], 0, 0);
    }
    WAIT_ASYNC0();
#else
    for (int w = f; w < W32P; w += 256)
        smask[w] = mask[(size_t)i * W32P + w];
#endif
    __syncthreads();

    const float di = dinv[i];
    float acc = di * H[(size_t)i * FDIM + f];       // self-loop base term
    for (int w = 0; w < W32P; ++w) {
        unsigned bits = smask[w];
        while (bits) {
            int j = (w << 5) + __builtin_ctz(bits);
            bits &= bits - 1;
            acc += dinv[j] * H[(size_t)j * FDIM + f];  // coalesced 1KB gather
        }
    }
    out[(size_t)i * FDIM + f] = fmaxf(di * acc, 0.0f);
}

// ---------------------------------------------------------------------------
extern "C" void kernel_launch(void* const* d_in, const int* in_sizes, int n_in,
                              void* d_out, int out_size, void* d_ws, size_t ws_size,
                              hipStream_t stream) {
    const float* X  = (const float*)d_in[0];        // [10000,256]
    const float* W  = (const float*)d_in[1];        // [256,256]
    const int* edges = (const int*)d_in[2];         // [2,320000]
    const int* src = edges;
    const int* tgt = edges + N_EDGES;
    float* out = (float*)d_out;

    // workspace carve-up (256B-aligned offsets)
    char* ws = (char*)d_ws;
    unsigned* mask = (unsigned*)ws;                       // 12,800,000 B
    unsigned* deg  = (unsigned*)(ws + 12800000);          //     40,000 B
    float*    dinv = (float*)   (ws + 12840192);          //     40,000 B
    float*    H    = (float*)   (ws + 12880384);          // 10,240,000 B

    // zero mask+deg (contiguous) every call -> deterministic
    (void)hipMemsetAsync(mask, 0, 12800000 + N_NODES * sizeof(unsigned), stream);

    gcn_build_adj<<<(N_EDGES + 255) / 256, 256, 0, stream>>>(src, tgt, mask, deg);
    gcn_dinv<<<(N_NODES + 255) / 256, 256, 0, stream>>>(deg, dinv);
    gcn_gemm_wmma<<<625, 512, 0, stream>>>(X, W, H);
    gcn_aggregate<<<N_NODES, FDIM, 0, stream>>>(H, mask, dinv, out);

    (void)in_sizes; (void)n_in; (void)out_size; (void)ws_size;
}